// TensorBasisLayer_46024869543996
// MI455X (gfx1250) — compile-verified
//
#include <hip/hip_runtime.h>
#include <math.h>

// ---------------------------------------------------------------------------
// TensorBasisLayer for MI455X (gfx1250).
// Phase 1: per-edge radial basis  u(d) * NORM_CONST * norm[l,r] * j_l(d*z[l,r])
//          -> d_ws as 100k rows of 48 floats (42 used, padded to 192B).
// Phase 2: per-quad real spherical harmonics (49 cols) in LDS, gathered rbf
//          rows staged into LDS via GLOBAL_LOAD_ASYNC_TO_LDS_B128 (ASYNCcnt),
//          then 470MB of coalesced b64 nontemporal streaming stores.
// ---------------------------------------------------------------------------

#ifndef __has_builtin
#define __has_builtin(x) 0
#endif

#if defined(__HIP_DEVICE_COMPILE__) && \
    __has_builtin(__builtin_amdgcn_global_load_async_to_lds_b128)
#define USE_ASYNC_COPY 1
#else
#define USE_ASYNC_COPY 0
#endif

#define AS_GLOBAL __attribute__((address_space(1)))
#define AS_LOCAL  __attribute__((address_space(3)))

typedef int   v4i __attribute__((vector_size(4 * sizeof(int))));
typedef float vf2 __attribute__((ext_vector_type(2)));
typedef int   vi2 __attribute__((ext_vector_type(2)));

#define S_DIM 7
#define R_DIM 6
#define SR    42
#define ROWP  48          // padded ws row stride (floats), 192B = 12 x b128
#define NSPH  49
#define NCOL  294         // 49 * 6
#define NPAIR 147         // NCOL / 2 (rows start even -> pairs never split rows)
#define QPB   128         // quads per block (= blockDim), 4 waves
#define RBF_STRIDE 52     // LDS row stride for rbf (16B aligned)
#define INV_CUTOFF 0.2f
#define NORM_CONST 0.08944271909999159f   // (1/5)^1.5

// First 6 zeros of spherical Bessel j_l, l = 0..6
__device__ __constant__ float g_ZER[S_DIM][R_DIM] = {
  { 3.14159265f,  6.28318531f,  9.42477796f, 12.56637061f, 15.70796327f, 18.84955592f},
  { 4.49340946f,  7.72525184f, 10.90412166f, 14.06619391f, 17.22075527f, 20.37130296f},
  { 5.76345920f,  9.09501133f, 12.32294097f, 15.51460301f, 18.68903636f, 21.85387422f},
  { 6.98793200f, 10.41711855f, 13.69802315f, 16.92362129f, 20.12180617f, 23.30424699f},
  { 8.18256145f, 11.70490715f, 15.03966471f, 18.30125596f, 21.52541773f, 24.72756555f},
  { 9.35581211f, 12.96653017f, 16.35470964f, 19.65315217f, 22.90455062f, 26.12775035f},
  {10.51283541f, 14.20739213f, 17.64797495f, 20.98346343f, 24.26276804f, 27.50786838f}};

// COEF[l][0] = sqrt((2l+1)/(4pi)); COEF[l][m>0] = sqrt((2l+1)/(2pi)*(l-m)!/(l+m)!)
__device__ __constant__ float g_COEF[S_DIM][S_DIM] = {
  {0.28209479f, 0.f,         0.f,         0.f,         0.f,         0.f,         0.f},
  {0.48860251f, 0.48860251f, 0.f,         0.f,         0.f,         0.f,         0.f},
  {0.63078313f, 0.36418281f, 0.18209141f, 0.f,         0.f,         0.f,         0.f},
  {0.74635267f, 0.30469720f, 0.09635369f, 0.03933623f, 0.f,         0.f,         0.f},
  {0.84628438f, 0.26761862f, 0.06307831f, 0.01685838f, 0.00596034f, 0.f,         0.f},
  {0.93560258f, 0.24157161f, 0.04565273f, 0.00931882f, 0.00219651f, 0.00069460f, 0.f},
  {1.01710723f, 0.22195093f, 0.03509366f, 0.00584893f, 0.00106787f, 0.00022767f, 0.00006572f}};

// Upward recurrence, identical formulation to the reference (f32).
__device__ __forceinline__ float sph_jl(int l, float z) {
  float s, c;
  sincosf(z, &s, &c);
  float j0 = s / z;
  if (l == 0) return j0;
  float j1 = s / (z * z) - c / z;
  for (int n = 1; n < l; ++n) {
    float jn = (2.0f * n + 1.0f) / z * j1 - j0;
    j0 = j1; j1 = jn;
  }
  return j1;
}

// ---------------------------------------------------------------------------
// Kernel 1: compact radial table, one edge per thread.
// ---------------------------------------------------------------------------
__global__ __launch_bounds__(256) void rbf_env_kernel(
    const float* __restrict__ D, float* __restrict__ ws, int ne) {
  __shared__ float nrmS[SR];
  const int t = threadIdx.x;
  if (t < SR) {  // sqrt(2)/|j_{l+1}(z_lr)| * NORM_CONST, computed once per block
    int l = t / R_DIM, r = t - (t / R_DIM) * R_DIM;
    nrmS[t] = 1.41421356237f * NORM_CONST / fabsf(sph_jl(l + 1, g_ZER[l][r]));
  }
  __syncthreads();

  int i = blockIdx.x * 256 + t;
  if (i >= ne) return;

  float d    = D[i] * INV_CUTOFF;
  float invd = 1.0f / d;
  float d2 = d * d, d4 = d2 * d2, d5 = d4 * d;
  // p=6 envelope: 1/d - 28 d^5 + 48 d^6 - 21 d^7, zero for d >= 1
  float env = invd + d5 * (-28.0f + d * (48.0f + d * (-21.0f)));
  float u = (d < 1.0f) ? env : 0.0f;

  float* row = ws + (size_t)i * ROWP;
#pragma unroll
  for (int l = 0; l < S_DIM; ++l) {
#pragma unroll
    for (int r = 0; r < R_DIM; ++r) {
      row[l * R_DIM + r] = u * nrmS[l * R_DIM + r] * sph_jl(l, d * g_ZER[l][r]);
    }
  }
}

// ---------------------------------------------------------------------------
// Kernel 2: sph harmonics + gather + 470MB streaming output.
// ---------------------------------------------------------------------------
__global__ __launch_bounds__(QPB) void tensor_basis_kernel(
    const float* __restrict__ alpha, const float* __restrict__ phi_in,
    const int* __restrict__ id4, const float* __restrict__ rbf,
    float* __restrict__ out, int nq) {
  __shared__ __align__(16) float rbfS[QPB * RBF_STRIDE];
  __shared__ float sphS[QPB * NSPH];
  __shared__ int   idS[QPB];
  __shared__ __align__(8) int tabS[NCOL];

  const int t  = threadIdx.x;
  const int q0 = blockIdx.x * QPB;
  const int qi = q0 + t;

  idS[t] = (qi < nq) ? id4[qi] : 0;
  __syncthreads();

  // --- stage 128 gathered rbf rows (192B each) into LDS via async DMA -----
#pragma unroll
  for (int k = 0; k < 12; ++k) {
    int chunk = k * QPB + t;              // 0..1535: 12 x b128 per row
    int row   = chunk / 12;
    int part  = chunk - row * 12;
    const float* src = rbf + (size_t)idS[row] * ROWP + part * 4;
    float*       dst = &rbfS[row * RBF_STRIDE + part * 4];
#if USE_ASYNC_COPY
    __builtin_amdgcn_global_load_async_to_lds_b128(
        (AS_GLOBAL v4i*)src, (AS_LOCAL v4i*)dst, 0, 0);
#else
    *(float4*)dst = *(const float4*)src;
#endif
  }

  // --- per-thread spherical harmonics (overlaps the async copies) ---------
  {
    float a  = (qi < nq) ? alpha[qi]  : 0.0f;
    float ph = (qi < nq) ? phi_in[qi] : 0.0f;
    float st, ct;
    sincosf(a, &st, &ct);

    float P[S_DIM][S_DIM];
    P[0][0] = 1.0f;
#pragma unroll
    for (int m = 1; m < S_DIM; ++m)
      P[m][m] = (float)(-(2 * m - 1)) * st * P[m - 1][m - 1];
#pragma unroll
    for (int m = 0; m + 1 < S_DIM; ++m)
      P[m + 1][m] = (float)(2 * m + 1) * ct * P[m][m];
#pragma unroll
    for (int m = 0; m < S_DIM; ++m) {
#pragma unroll
      for (int l = m + 2; l < S_DIM; ++l)
        P[l][m] = ((float)(2 * l - 1) * ct * P[l - 1][m] -
                   (float)(l + m - 1) * P[l - 2][m]) / (float)(l - m);
    }

    float sm[S_DIM], cm[S_DIM];
    sm[0] = 0.0f; cm[0] = 1.0f;
#pragma unroll
    for (int m = 1; m < S_DIM; ++m) sincosf((float)m * ph, &sm[m], &cm[m]);

    float* myS = &sphS[t * NSPH];
#pragma unroll
    for (int l = 0; l < S_DIM; ++l) {
      myS[l * l] = g_COEF[l][0] * P[l][0];
#pragma unroll
      for (int m = 1; m <= l; ++m) {
        float v = g_COEF[l][m] * P[l][m];
        myS[l * l + m]             = v * sm[m];   // sin terms, m ascending
        myS[l * l + 2 * l + 1 - m] = v * cm[m];   // cos terms, m descending
      }
    }
  }

  // --- column lookup: col -> (sph index q | rbf offset l*6+r) -------------
  for (int e = t; e < NCOL; e += QPB) {
    int q = e / 6;
    int r = e - q * 6;
    int l = (int)sqrtf((float)q);            // block l starts at l*l
    tabS[e] = q | ((l * 6 + r) << 8);
  }

#if USE_ASYNC_COPY
#if __has_builtin(__builtin_amdgcn_s_wait_asynccnt)
  __builtin_amdgcn_s_wait_asynccnt(0);
#else
  asm volatile("s_wait_asynccnt 0" ::: "memory");
#endif
#endif
  __syncthreads();

  // --- streaming output: b64 NT stores, pair index = t + 128k -------------
  // Rows are 294 elements (even) starting at even offsets, so an element
  // pair never straddles a row; row advances by 0/1 per step (128 < 147).
  vf2* __restrict__ outB = (vf2*)(out + (long long)q0 * NCOL);
  int row = 0, rem2 = t, idx2 = t;             // t < 128 < 147 -> row 0
  const bool full = (q0 + QPB) <= nq;
  if (full) {
    for (int k = 0; k < NPAIR; ++k) {
      vi2 info = *(const vi2*)&tabS[2 * rem2];
      vf2 v;
      v.x = sphS[row * NSPH + (info.x & 0xFF)] *
            rbfS[row * RBF_STRIDE + (info.x >> 8)];
      v.y = sphS[row * NSPH + (info.y & 0xFF)] *
            rbfS[row * RBF_STRIDE + (info.y >> 8)];
      __builtin_nontemporal_store(v, outB + idx2);
      idx2 += QPB;
      rem2 += QPB;
      if (rem2 >= NPAIR) { rem2 -= NPAIR; ++row; }
    }
  } else {
    for (int k = 0; k < NPAIR; ++k) {
      if (q0 + row < nq) {
        vi2 info = *(const vi2*)&tabS[2 * rem2];
        vf2 v;
        v.x = sphS[row * NSPH + (info.x & 0xFF)] *
              rbfS[row * RBF_STRIDE + (info.x >> 8)];
        v.y = sphS[row * NSPH + (info.y & 0xFF)] *
              rbfS[row * RBF_STRIDE + (info.y >> 8)];
        __builtin_nontemporal_store(v, outB + idx2);
      }
      idx2 += QPB;
      rem2 += QPB;
      if (rem2 >= NPAIR) { rem2 -= NPAIR; ++row; }
    }
  }
}

// ---------------------------------------------------------------------------
extern "C" void kernel_launch(void* const* d_in, const int* in_sizes, int n_in,
                              void* d_out, int out_size, void* d_ws, size_t ws_size,
                              hipStream_t stream) {
  (void)n_in; (void)out_size; (void)ws_size;
  const float* D     = (const float*)d_in[0];
  const float* Alpha = (const float*)d_in[1];
  const float* Theta = (const float*)d_in[2];
  const int*   id4   = (const int*)d_in[3];
  // d_in[4] (Kidx) is unused by the reference computation.
  float* out = (float*)d_out;
  float* ws  = (float*)d_ws;          // needs ne * 48 * 4 = 19.2 MB
  const int ne = in_sizes[0];
  const int nq = in_sizes[1];

  rbf_env_kernel<<<(ne + 255) / 256, 256, 0, stream>>>(D, ws, ne);
  tensor_basis_kernel<<<(nq + QPB - 1) / QPB, QPB, 0, stream>>>(
      Alpha, Theta, id4, ws, out, nq);
}